// GraphConvolution_66554813218924
// MI455X (gfx1250) — compile-verified
//
#include <hip/hip_runtime.h>
#include <hip/hip_bf16.h>

typedef __attribute__((ext_vector_type(2))) float v2f;
typedef __attribute__((ext_vector_type(8))) float v8f;

#define D_FEAT 256
#define UNITS  256

// ---------------------------------------------------------------------------
// Phase 0b: repack W[256][256] into pair-interleaved Wp[128][256][2]:
//   Wp[kp][n] = { W[2*kp][n], W[2*kp+1][n] }
// so a B fragment (VGPR0=W[k][n], VGPR1=W[k+1][n]) is ONE b64 load.
// ---------------------------------------------------------------------------
__global__ __launch_bounds__(256) void gcn_pack_w(
    const float* __restrict__ W,   // [256,256]
    float*       __restrict__ Wp)  // [128,256,2]
{
    int idx = blockIdx.x * 256 + threadIdx.x;   // 0 .. 128*256-1
    int kp = idx >> 8;                          // 0..127
    int n  = idx & 255;
    v2f p;
    p.x = W[(2 * kp)     * UNITS + n];
    p.y = W[(2 * kp + 1) * UNITS + n];
    *reinterpret_cast<v2f*>(Wp + (size_t)idx * 2) = p;
}

// ---------------------------------------------------------------------------
// Phase 1: SpMM scatter. agg[row[e]] += val[e] * features[col[e]]
// 64 threads (2 waves) cooperate on one edge; each thread owns one float4
// quad of the 256-wide feature row. Uses hardware global_atomic_add_f32.
// agg buffer is L2-resident (102.4 MB < 192 MB L2).
// ---------------------------------------------------------------------------
__global__ __launch_bounds__(256) void gcn_spmm_scatter(
    const float* __restrict__ feat,   // [N_nodes, 256]
    const int*   __restrict__ erow,   // [E]
    const int*   __restrict__ ecol,   // [E]
    const float* __restrict__ eval_,  // [E]
    float*       __restrict__ agg,    // [N_nodes, 256] (pre-zeroed)
    int E)
{
    int gid = blockIdx.x * 256 + threadIdx.x;
    int e = gid >> 6;              // edge index (64 threads per edge)
    if (e >= E) return;
    int q = gid & 63;              // which float4 quad of the feature row

    int r = erow[e];
    int c = ecol[e];
    float v = eval_[e];

    const float4* src = reinterpret_cast<const float4*>(feat + (size_t)c * D_FEAT);
    float4 f = src[q];

    float* dst = agg + (size_t)r * D_FEAT + q * 4;
    unsafeAtomicAdd(dst + 0, v * f.x);   // global_atomic_add_f32
    unsafeAtomicAdd(dst + 1, v * f.y);
    unsafeAtomicAdd(dst + 2, v * f.z);
    unsafeAtomicAdd(dst + 3, v * f.w);
}

// ---------------------------------------------------------------------------
// Phase 2: out = relu(agg @ W + bias) using V_WMMA_F32_16X16X4_F32 (exact f32).
// Each wave computes a 16x64 strip of C (4 accumulators); 8 waves per block
// cover a 128-row slab. K=256 -> 64 WMMA steps; per step: 1 b64 A-load,
// 4 b64 B-loads (pair-packed Wp), 4 WMMAs.
// ---------------------------------------------------------------------------
__global__ __launch_bounds__(256) void gcn_gemm_bias_relu(
    const float* __restrict__ A,     // [M, 256]  aggregated features
    const float* __restrict__ Wp,    // [128, 256, 2] pair-packed weights
    const float* __restrict__ bias,  // [256]
    float*       __restrict__ out,   // [M, 256]
    int M)
{
    constexpr int K = D_FEAT;
    constexpr int N = UNITS;

    const int wave = threadIdx.x >> 5;        // 0..7
    const int lane = threadIdx.x & 31;
    const int m0   = (blockIdx.x * 8 + wave) * 16;
    if (m0 >= M) return;                       // wave-uniform: EXEC stays all-1s
    const int n0   = blockIdx.y * 64;

    const int mrow = lane & 15;                // A-fragment: M = lane%16
    const int half = lane >> 4;                // 0: K pair {0,1}; 1: K pair {2,3}
    const int ncol = lane & 15;                // B/C-fragment: N = lane%16

    // per-lane bias for the 4 N-tiles
    const float b0 = bias[n0 + 0  + ncol];
    const float b1 = bias[n0 + 16 + ncol];
    const float b2 = bias[n0 + 32 + ncol];
    const float b3 = bias[n0 + 48 + ncol];

    v8f acc0 = {}, acc1 = {}, acc2 = {}, acc3 = {};

    const float* Arow = A + (size_t)(m0 + mrow) * K;
    const v2f*   Wp2  = reinterpret_cast<const v2f*>(Wp);   // [128][256] of pairs

    #pragma unroll 4
    for (int k0 = 0; k0 < K; k0 += 4) {
        // A 16x4 fragment: VGPR0 = A[m, k0+2*half], VGPR1 = A[m, k0+2*half+1]
        v2f a = *reinterpret_cast<const v2f*>(Arow + k0 + 2 * half);

        // B 4x16 fragments: one b64 pair-load each
        const v2f* wk = Wp2 + (size_t)(k0 / 2 + half) * N + n0 + ncol;
        v2f fb0 = wk[0];
        v2f fb1 = wk[16];
        v2f fb2 = wk[32];
        v2f fb3 = wk[48];

        acc0 = __builtin_amdgcn_wmma_f32_16x16x4_f32(false, a, false, fb0,
                                                     (short)0, acc0, false, false);
        acc1 = __builtin_amdgcn_wmma_f32_16x16x4_f32(false, a, false, fb1,
                                                     (short)0, acc1, false, false);
        acc2 = __builtin_amdgcn_wmma_f32_16x16x4_f32(false, a, false, fb2,
                                                     (short)0, acc2, false, false);
        acc3 = __builtin_amdgcn_wmma_f32_16x16x4_f32(false, a, false, fb3,
                                                     (short)0, acc3, false, false);
    }

    // Epilogue: C/D layout — VGPR r holds M = m0 + r + 8*half, N = n0 + tile*16 + ncol
    #pragma unroll
    for (int r = 0; r < 8; ++r) {
        const size_t row = (size_t)(m0 + r + 8 * half) * N;
        float x0 = acc0[r] + b0;
        float x1 = acc1[r] + b1;
        float x2 = acc2[r] + b2;
        float x3 = acc3[r] + b3;
        out[row + n0 + 0  + ncol] = x0 > 0.0f ? x0 : 0.0f;
        out[row + n0 + 16 + ncol] = x1 > 0.0f ? x1 : 0.0f;
        out[row + n0 + 32 + ncol] = x2 > 0.0f ? x2 : 0.0f;
        out[row + n0 + 48 + ncol] = x3 > 0.0f ? x3 : 0.0f;
    }
}

extern "C" void kernel_launch(void* const* d_in, const int* in_sizes, int n_in,
                              void* d_out, int out_size, void* d_ws, size_t ws_size,
                              hipStream_t stream) {
    const float* features = (const float*)d_in[0];   // [N_nodes, 256]
    const int*   erow     = (const int*)  d_in[1];   // [E]
    const int*   ecol     = (const int*)  d_in[2];   // [E]
    const float* eval_    = (const float*)d_in[3];   // [E]
    const float* kern     = (const float*)d_in[4];   // [256, 256]
    const float* bias     = (const float*)d_in[5];   // [256]
    float*       out      = (float*)d_out;

    const int E       = in_sizes[1];
    const int n_nodes = in_sizes[0] / D_FEAT;        // 100000

    // workspace layout: [agg: n_nodes*256 f32][Wp: 128*256*2 f32]
    float* agg = (float*)d_ws;
    const size_t agg_bytes = (size_t)n_nodes * D_FEAT * sizeof(float);
    float* Wp  = (float*)((char*)d_ws + agg_bytes);

    // zero the aggregation buffer (memset node is graph-capture safe)
    hipMemsetAsync(agg, 0, agg_bytes, stream);

    // Phase 0b: pair-pack the weight matrix (128*256 pairs, 128 blocks)
    gcn_pack_w<<<(D_FEAT / 2) * UNITS / 256, 256, 0, stream>>>(kern, Wp);

    // Phase 1: edge scatter (E * 64 work-items)
    {
        long long work = (long long)E * 64;
        int blocks = (int)((work + 255) / 256);
        gcn_spmm_scatter<<<blocks, 256, 0, stream>>>(features, erow, ecol, eval_,
                                                     agg, E);
    }

    // Phase 2: WMMA GEMM + bias + relu
    {
        dim3 grid((n_nodes + 127) / 128, UNITS / 64);
        gcn_gemm_bias_relu<<<grid, 256, 0, stream>>>(agg, Wp, bias, out, n_nodes);
    }
}